// UpSample_46136538694253
// MI455X (gfx1250) — compile-verified
//
#include <hip/hip_runtime.h>

// ---------------------------------------------------------------------------
// 3-NN feature upsampling (PointNet++-style) for MI455X / gfx1250.
//
// Roofline: ~40 MB HBM traffic (out 32MB streamed + x1 8MB, L2-resident)
// @ 23.3 TB/s => ~1.7us floor. Distance tiles via V_WMMA_F32_16X16X4_F32
// (exact f32). Top-3 selection is lane-local over packed {dist|idx} keys
// using a branch-free min/max network (no VCC/cndmask chains).
// ---------------------------------------------------------------------------

typedef float v2f __attribute__((ext_vector_type(2)));
typedef float v8f __attribute__((ext_vector_type(8)));

namespace {
constexpr int B  = 8;
constexpr int N1 = 1024;   // coarse points
constexpr int N2 = 4096;   // fine points
constexpr int C  = 256;    // channels
constexpr int PTS_PER_BLOCK = 128;   // fine points per workgroup (8 waves x 16)
constexpr int THREADS = 256;         // 8 wave32
constexpr float EPSILON = 1e-8f;
constexpr int IDX_MASK = N1 - 1;     // 10 index bits packed into the key LSBs
}

__global__ __launch_bounds__(THREADS)
void knn3_upsample_kernel(const float* __restrict__ p1,   // [B, N1, 3]
                          const float* __restrict__ x1,   // [B, C, N1]
                          const float* __restrict__ p2,   // [B, N2, 3]
                          float* __restrict__ out)        // [B, C, N2]
{
    // Preprocessed coarse points: (-2x, -2y, -2z, |p|^2)  -> 16 KB
    __shared__ float4 s_p1[N1];
    __shared__ float  s_w[3][PTS_PER_BLOCK];
    __shared__ int    s_i[3][PTS_PER_BLOCK];

    const int bb  = blockIdx.x / (N2 / PTS_PER_BLOCK);   // batch
    const int blk = blockIdx.x % (N2 / PTS_PER_BLOCK);   // fine-point window
    const int tid = threadIdx.x;

    // ---- phase 0: stage preprocessed coarse points into LDS ---------------
    {
        const float* p1b = p1 + (size_t)bb * N1 * 3;
        for (int p = tid; p < N1; p += THREADS) {
            float x = p1b[p * 3 + 0];
            float y = p1b[p * 3 + 1];
            float z = p1b[p * 3 + 2];
            s_p1[p] = make_float4(-2.0f * x, -2.0f * y, -2.0f * z,
                                  x * x + y * y + z * z);
        }
    }
    __syncthreads();

    // ---- phase 1: per-wave 3-NN over all coarse tiles via f32 WMMA --------
    const int wave = tid >> 5;
    const int lane = tid & 31;
    const int col  = lane & 15;       // fine point within the wave's 16
    const int half = lane >> 4;       // 0: lanes 0-15, 1: lanes 16-31
    const int f    = blk * PTS_PER_BLOCK + wave * 16 + col;

    // B matrix (4x16): K rows = (x, y, z, 1) of the 16 fine points.
    // Lanes 0-15 hold K0/K1, lanes 16-31 hold K2/K3 (mirrors the A layout).
    const float* p2f = p2 + ((size_t)bb * N2 + f) * 3;
    const float fx = p2f[0], fy = p2f[1], fz = p2f[2];
    const float pn2 = fx * fx + fy * fy + fz * fz;

    v2f bmat;
    bmat.x = half ? fz   : fx;   // K2 : K0
    bmat.y = half ? 1.0f : fy;   // K3 : K1

    // Sorted top-3 keys (ascending). Key = {dist_bits[31:10] | coarse_idx}.
    // Distances are biased by |p2|^2 so they are >= -epsilon_round; a tiny
    // rounding-negative value is genuinely the nearest point and signed-int
    // ordering still ranks it first.
    int k0 = 0x7fffffff, k1 = 0x7fffffff, k2 = 0x7fffffff;

    auto insert3 = [&](int key) {
        int t = min(key, k2);            // candidate enters the window
        k2 = max(t, k1);
        k1 = min(max(t, k0), k1);        // med3(t, k0, k1)
        k0 = min(t, k0);
    };

    // A matrix (16x4 f32, 2 VGPRs): lanes 0-15 read (K0,K1)=(-2x,-2y),
    // lanes 16-31 read (K2,K3)=(-2z,|p1|^2) of coarse point M = t*16+col.
    const float* lds_f   = (const float*)&s_p1[0] + half * 2;
    const int    halfoff = half * 8;

    #pragma unroll 4
    for (int t = 0; t < N1 / 16; ++t) {
        float2 av = *(const float2*)(lds_f + (size_t)(t * 16 + col) * 4);
        v2f amat; amat.x = av.x; amat.y = av.y;

        v8f acc = {};
        acc = __builtin_amdgcn_wmma_f32_16x16x4_f32(
                  /*neg_a=*/false, amat, /*neg_b=*/false, bmat,
                  /*c_mod=*/(short)0, acc, /*reuse_a=*/false, /*reuse_b=*/false);

        // D layout: VGPR r, lanes 0-15 -> row M=r; lanes 16-31 -> row M=r+8.
        const int ibase = t * 16 + halfoff;
        #pragma unroll
        for (int r = 0; r < 8; ++r) {
            float v  = acc[r] + pn2;
            int  key = (__float_as_int(v) & ~IDX_MASK) | (ibase + r);
            insert3(key);
        }
    }

    // Merge the two half-wave partial top-3s (index rides inside the key).
    {
        int e0 = __shfl_xor(k0, 16);
        int e1 = __shfl_xor(k1, 16);
        int e2 = __shfl_xor(k2, 16);
        insert3(e0);
        insert3(e1);
        insert3(e2);
    }

    // Recover indices, recompute exact f32 distances from the LDS table,
    // and form normalized reciprocal-distance weights.
    {
        const int i0 = k0 & IDX_MASK, i1 = k1 & IDX_MASK, i2 = k2 & IDX_MASK;
        float4 q0 = s_p1[i0], q1 = s_p1[i1], q2 = s_p1[i2];
        float d0 = fmaf(q0.x, fx, fmaf(q0.y, fy, fmaf(q0.z, fz, q0.w))) + pn2;
        float d1 = fmaf(q1.x, fx, fmaf(q1.y, fy, fmaf(q1.z, fz, q1.w))) + pn2;
        float d2 = fmaf(q2.x, fx, fmaf(q2.y, fy, fmaf(q2.z, fz, q2.w))) + pn2;
        float r0 = 1.0f / (d0 + EPSILON);
        float r1 = 1.0f / (d1 + EPSILON);
        float r2 = 1.0f / (d2 + EPSILON);
        float inv = 1.0f / (r0 + r1 + r2);
        if (half == 0) {
            int slot = wave * 16 + col;
            s_w[0][slot] = r0 * inv;  s_i[0][slot] = i0;
            s_w[1][slot] = r1 * inv;  s_i[1][slot] = i1;
            s_w[2][slot] = r2 * inv;  s_i[2][slot] = i2;
        }
    }
    __syncthreads();

    // ---- phase 2: cooperative gather + blend over all 256 channels --------
    // Thread owns one fine point (coalesced N2-contiguous stores) and walks
    // channels with stride 2. x1 (8 MB/dispatch) stays hot in the 192 MB L2;
    // output is write-once => non-temporal stores keep it from evicting x1.
    {
        const int   mloc = tid & (PTS_PER_BLOCK - 1);
        const int   c0   = tid >> 7;   // 0 or 1
        const float w0 = s_w[0][mloc], w1 = s_w[1][mloc], w2 = s_w[2][mloc];
        const int   a0 = s_i[0][mloc], a1 = s_i[1][mloc], a2 = s_i[2][mloc];

        const float* xb = x1 + (size_t)bb * C * N1;
        float* ob = out + (size_t)bb * C * N2 + (size_t)blk * PTS_PER_BLOCK + mloc;

        #pragma unroll 4
        for (int c = c0; c < C; c += 2) {
            const float* xr = xb + (size_t)c * N1;
            float v = w0 * xr[a0] + w1 * xr[a1] + w2 * xr[a2];
            __builtin_nontemporal_store(v, ob + (size_t)c * N2);
        }
    }
}

extern "C" void kernel_launch(void* const* d_in, const int* in_sizes, int n_in,
                              void* d_out, int out_size, void* d_ws, size_t ws_size,
                              hipStream_t stream) {
    (void)in_sizes; (void)n_in; (void)out_size; (void)d_ws; (void)ws_size;
    const float* p1 = (const float*)d_in[0];   // [B, N1, 3]
    const float* x1 = (const float*)d_in[1];   // [B, C, N1]
    const float* p2 = (const float*)d_in[2];   // [B, N2, 3]
    float* out = (float*)d_out;                // [B, C, N2]

    dim3 grid(B * (N2 / PTS_PER_BLOCK));       // 8 * 32 = 256 blocks
    knn3_upsample_kernel<<<grid, THREADS, 0, stream>>>(p1, x1, p2, out);
}